// EquivariantInteractionBlock_69965017252203
// MI455X (gfx1250) — compile-verified
//
#include <hip/hip_runtime.h>

// ---------------------------------------------------------------------------
// Equivariant interaction block, fused for MI455X (gfx1250, wave32, WMMA).
//   MUL0=32, MUL1=16, RBF=16, HID=64, WNUMEL=2304
//   O1=1024, O2=1536, O3=1792
// ---------------------------------------------------------------------------

typedef __attribute__((ext_vector_type(2))) float v2f;
typedef __attribute__((ext_vector_type(8))) float v8f;

__device__ __forceinline__ v8f wmma4(v2f a, v2f b, v8f c) {
  // V_WMMA_F32_16X16X4_F32 : D = A(16x4) * B(4x16) + C(16x16), exact fp32
  return __builtin_amdgcn_wmma_f32_16x16x4_f32(false, a, false, b, (short)0, c,
                                               false, false);
}

__device__ __forceinline__ float silu_f(float v) {
  // v_exp_f32 + v_rcp_f32 (native, ~1ulp) instead of IEEE div expansion
  return v * __builtin_amdgcn_rcpf(1.0f + __expf(-v));
}

#define C_PATH 0.14433756729740643f /* 1/sqrt(48) */
#define C_110 0.5773502691896258f   /* 1/sqrt(3)  */

// ---------------------------------------------------------------------------
// Kernel 0: zero agg (N*80) + deg (N) in workspace
// ---------------------------------------------------------------------------
__global__ void zero_ws_kernel(float* __restrict__ p, long n) {
  long i = (long)blockIdx.x * blockDim.x + threadIdx.x;
  long stride = (long)gridDim.x * blockDim.x;
  for (; i < n; i += stride) p[i] = 0.0f;
}

// ---------------------------------------------------------------------------
// Kernel 1: per-edge fused MLP (WMMA) + tensor product + scatter
// One workgroup (256 thr = 8 waves) handles 16 edges.
// ---------------------------------------------------------------------------
__global__ __launch_bounds__(256) void edge_kernel(
    const float* __restrict__ x, const int* __restrict__ esrc,
    const int* __restrict__ edst, const float* __restrict__ esh,
    const float* __restrict__ erbf, const float* __restrict__ w1,
    const float* __restrict__ b1, const float* __restrict__ w2,
    const float* __restrict__ b2, const float* __restrict__ w3,
    const float* __restrict__ b3, float* __restrict__ agg,
    float* __restrict__ deg, int E) {
  __shared__ float sH1[16 * 66];  // silu(rbf@w1+b1), row stride 66 (bank pad)
  __shared__ float sH2[16 * 66];  // silu(h1@w2+b2)
  __shared__ float sX[16 * 80];   // gathered x[src]
  __shared__ float sSH[16 * 4];   // edge_sh (zeroed for invalid edges)
  __shared__ float sCS[16 * 32];  // xs * sh_s        (W1 coeff)
  __shared__ float sDV[16 * 16];  // C110 * xv.sh_v   (W4 coeff)
  __shared__ float sACC[16 * 80]; // per-edge message accumulator
  __shared__ int sSRC[16];
  __shared__ int sDST[16];

  const int t = threadIdx.x;
  const int wid = t >> 5;
  const int lane = t & 31;
  const int n16 = lane & 15;
  const int hi = lane >> 4;  // 0: K=k0,k0+1 / rows 0..7 ; 1: K=k0+2,k0+3 / rows 8..15
  const int e0 = blockIdx.x * 16;

  if (t < 16) {
    int eg = min(e0 + t, E - 1);
    sSRC[t] = esrc[eg];
    sDST[t] = edst[eg];
    bool valid = (e0 + t) < E;
#pragma unroll
    for (int d = 0; d < 4; ++d) sSH[t * 4 + d] = valid ? esh[eg * 4 + d] : 0.0f;
  }
  __syncthreads();

  // gather source features, zero accumulator
  for (int idx = t; idx < 16 * 80; idx += 256) {
    int e = idx / 80;
    int c = idx - e * 80;
    sX[idx] = x[(size_t)sSRC[e] * 80 + c];
    sACC[idx] = 0.0f;
  }
  __syncthreads();

  // TP coefficients
  for (int idx = t; idx < 16 * 32; idx += 256) {
    int e = idx >> 5, i = idx & 31;
    sCS[idx] = sX[e * 80 + i] * sSH[e * 4 + 0];
  }
  for (int idx = t; idx < 16 * 16; idx += 256) {
    int e = idx >> 4, i = idx & 15;
    float dv = 0.0f;
#pragma unroll
    for (int d = 0; d < 3; ++d)
      dv += sX[e * 80 + 32 + i * 3 + d] * sSH[e * 4 + 1 + d];
    sDV[idx] = C_110 * dv;
  }

  // ---- layer 1: h1 = silu(rbf @ w1 + b1), waves 0..3 each own one N-tile ----
  if (wid < 4) {
    const int nt = wid;
    v8f c = {};
    const int mg = min(e0 + n16, E - 1);
#pragma unroll
    for (int kk = 0; kk < 4; ++kk) {
      const int k0 = kk * 4 + hi * 2;
      float2 av = *(const float2*)&erbf[(size_t)mg * 16 + k0];
      v2f a;
      a[0] = av.x;
      a[1] = av.y;
      v2f b;
      b[0] = w1[k0 * 64 + nt * 16 + n16];
      b[1] = w1[(k0 + 1) * 64 + nt * 16 + n16];
      c = wmma4(a, b, c);
    }
#pragma unroll
    for (int v = 0; v < 8; ++v) {
      const int m = v + hi * 8;
      const int col = nt * 16 + n16;
      sH1[m * 66 + col] = silu_f(c[v] + b1[col]);
    }
  }
  __syncthreads();

  // ---- layer 2: h2 = silu(h1 @ w2 + b2) ----
  if (wid < 4) {
    const int nt = wid;
    v8f c = {};
#pragma unroll
    for (int kk = 0; kk < 16; ++kk) {
      const int k0 = kk * 4 + hi * 2;
      float2 av = *(const float2*)&sH1[n16 * 66 + k0];
      v2f a;
      a[0] = av.x;
      a[1] = av.y;
      v2f b;
      b[0] = w2[k0 * 64 + nt * 16 + n16];
      b[1] = w2[(k0 + 1) * 64 + nt * 16 + n16];
      c = wmma4(a, b, c);
    }
#pragma unroll
    for (int v = 0; v < 8; ++v) {
      const int m = v + hi * 8;
      const int col = nt * 16 + n16;
      sH2[m * 66 + col] = silu_f(c[v] + b2[col]);
    }
  }
  __syncthreads();

  // ---- layer 3 fused with tensor product: stream 144 N-tiles of W ----
  // Region boundaries (1024/1536/1792) are multiples of 16 -> wave-uniform.
  for (int j = wid; j < 144; j += 8) {
    const int cb = j * 16;
    v8f c = {};
#pragma unroll
    for (int kk = 0; kk < 16; ++kk) {
      const int k0 = kk * 4 + hi * 2;
      float2 av = *(const float2*)&sH2[n16 * 66 + k0];
      v2f a;
      a[0] = av.x;
      a[1] = av.y;
      v2f b;
      b[0] = w3[k0 * 2304 + cb + n16];
      b[1] = w3[(k0 + 1) * 2304 + cb + n16];
      c = wmma4(a, b, c);
    }
    const int col = cb + n16;        // global W column for this lane
    const float bias = b3[col];
    if (cb < 1024) {                 // W1: out_s[o] += (xs[i]*sh_s) * W
      const int i = col >> 5, o = col & 31;
#pragma unroll
      for (int v = 0; v < 8; ++v) {
        const int e = v + hi * 8;
        atomicAdd(&sACC[e * 80 + o], sCS[e * 32 + i] * (c[v] + bias));
      }
    } else if (cb < 1536) {          // W2: out_v[o,d] += xs[i]*sh_v[d]*W
      const int rel = col - 1024, i = rel >> 4, o = rel & 15;
#pragma unroll
      for (int v = 0; v < 8; ++v) {
        const int e = v + hi * 8;
        const float tt = sX[e * 80 + i] * (c[v] + bias);
        atomicAdd(&sACC[e * 80 + 32 + o * 3 + 0], tt * sSH[e * 4 + 1]);
        atomicAdd(&sACC[e * 80 + 32 + o * 3 + 1], tt * sSH[e * 4 + 2]);
        atomicAdd(&sACC[e * 80 + 32 + o * 3 + 2], tt * sSH[e * 4 + 3]);
      }
    } else if (cb < 1792) {          // W3: out_v[o,d] += xv[i,d]*sh_s*W
      const int rel = col - 1536, i = rel >> 4, o = rel & 15;
#pragma unroll
      for (int v = 0; v < 8; ++v) {
        const int e = v + hi * 8;
        const float tt = sSH[e * 4 + 0] * (c[v] + bias);
        atomicAdd(&sACC[e * 80 + 32 + o * 3 + 0], tt * sX[e * 80 + 32 + i * 3 + 0]);
        atomicAdd(&sACC[e * 80 + 32 + o * 3 + 1], tt * sX[e * 80 + 32 + i * 3 + 1]);
        atomicAdd(&sACC[e * 80 + 32 + o * 3 + 2], tt * sX[e * 80 + 32 + i * 3 + 2]);
      }
    } else {                         // W4: out_s[o] += C110*(xv.sh_v)[i]*W
      const int rel = col - 1792, i = rel >> 5, o = rel & 31;
#pragma unroll
      for (int v = 0; v < 8; ++v) {
        const int e = v + hi * 8;
        atomicAdd(&sACC[e * 80 + o], sDV[e * 16 + i] * (c[v] + bias));
      }
    }
  }
  __syncthreads();

  // ---- scatter messages to destination nodes ----
  for (int idx = t; idx < 16 * 80; idx += 256) {
    const int e = idx / 80;
    if (e0 + e < E) {
      const int comp = idx - e * 80;
      atomicAdd(&agg[(size_t)sDST[e] * 80 + comp], C_PATH * sACC[idx]);
    }
  }
  if (t < 16 && (e0 + t) < E) atomicAdd(&deg[sDST[t]], 1.0f);
}

// ---------------------------------------------------------------------------
// Kernel 2: out = irrep_linear(x, self) + irrep_linear(agg/max(deg,1), out)
// ---------------------------------------------------------------------------
__global__ __launch_bounds__(256) void node_kernel(
    const float* __restrict__ x, const float* __restrict__ agg,
    const float* __restrict__ deg, const float* __restrict__ ws_self,
    const float* __restrict__ wv_self, const float* __restrict__ ws_out,
    const float* __restrict__ wv_out, float* __restrict__ out, int N) {
  __shared__ float sWsS[32 * 32], sWsO[32 * 32];
  __shared__ float sWvS[16 * 16], sWvO[16 * 16];
  for (int i = threadIdx.x; i < 1024; i += 256) {
    sWsS[i] = ws_self[i];
    sWsO[i] = ws_out[i];
  }
  for (int i = threadIdx.x; i < 256; i += 256) {
    sWvS[i] = wv_self[i];
    sWvO[i] = wv_out[i];
  }
  __syncthreads();

  const long total = (long)N * 80;
  const long stride = (long)gridDim.x * blockDim.x;
  for (long id = (long)blockIdx.x * blockDim.x + threadIdx.x; id < total;
       id += stride) {
    const int n = (int)(id / 80);
    const int comp = (int)(id - (long)n * 80);
    const float inv = __builtin_amdgcn_rcpf(fmaxf(deg[n], 1.0f));
    const float* xr = x + (size_t)n * 80;
    const float* ar = agg + (size_t)n * 80;
    float acc = 0.0f;
    if (comp < 32) {
      const int o = comp;
#pragma unroll
      for (int i = 0; i < 32; ++i)
        acc += xr[i] * sWsS[i * 32 + o] + (ar[i] * inv) * sWsO[i * 32 + o];
    } else {
      const int rem = comp - 32;
      const int o = rem / 3;
      const int d = rem - o * 3;
#pragma unroll
      for (int i = 0; i < 16; ++i)
        acc += xr[32 + i * 3 + d] * sWvS[i * 16 + o] +
               (ar[32 + i * 3 + d] * inv) * sWvO[i * 16 + o];
    }
    out[id] = acc;
  }
}

// ---------------------------------------------------------------------------
extern "C" void kernel_launch(void* const* d_in, const int* in_sizes, int n_in,
                              void* d_out, int out_size, void* d_ws,
                              size_t ws_size, hipStream_t stream) {
  (void)n_in;
  (void)out_size;
  (void)ws_size;
  const float* x = (const float*)d_in[0];
  const int* esrc = (const int*)d_in[1];
  const int* edst = (const int*)d_in[2];
  const float* esh = (const float*)d_in[3];
  const float* erbf = (const float*)d_in[4];
  const float* w1 = (const float*)d_in[5];
  const float* b1 = (const float*)d_in[6];
  const float* w2 = (const float*)d_in[7];
  const float* b2 = (const float*)d_in[8];
  const float* w3 = (const float*)d_in[9];
  const float* b3 = (const float*)d_in[10];
  const float* ws_self = (const float*)d_in[11];
  const float* wv_self = (const float*)d_in[12];
  const float* ws_out = (const float*)d_in[13];
  const float* wv_out = (const float*)d_in[14];

  const int N = in_sizes[0] / 80;
  const int E = in_sizes[1];

  float* agg = (float*)d_ws;            // N*80 floats
  float* deg = agg + (size_t)N * 80;    // N floats

  const long ztot = (long)N * 81;
  int zblocks = (int)((ztot + 255) / 256);
  if (zblocks > 4096) zblocks = 4096;
  zero_ws_kernel<<<zblocks, 256, 0, stream>>>(agg, ztot);

  const int tiles = (E + 15) / 16;
  edge_kernel<<<tiles, 256, 0, stream>>>(x, esrc, edst, esh, erbf, w1, b1, w2,
                                         b2, w3, b3, agg, deg, E);

  const long ntot = (long)N * 80;
  const int nblocks = (int)((ntot + 255) / 256);
  node_kernel<<<nblocks, 256, 0, stream>>>(x, agg, deg, ws_self, wv_self,
                                           ws_out, wv_out, (float*)d_out, N);
}